// COTREC_64991445123903
// MI455X (gfx1250) — compile-verified
//
#include <hip/hip_runtime.h>
#include <hip/hip_bf16.h>

typedef __attribute__((ext_vector_type(2))) float v2f;
typedef __attribute__((ext_vector_type(8))) float v8f;
typedef __attribute__((ext_vector_type(4))) int v4i;
typedef __attribute__((ext_vector_type(2))) int v2i;

#define BDIM   4096
#define MAXLEN 200
#define EE     112
#define LAYERS 3

#define TMROWS 32          // rows per block (2 strips x 16)
#define ASTRIDE 20         // LDS stride for A tile: 20r mod 64 injective, 16B-aligned
#define BSTRIDE 144        // LDS stride for B tile: halves land 32 banks apart, 16B-aligned
#define A_ELEMS (TMROWS * ASTRIDE)   // 640 floats
#define B_ELEMS (16 * BSTRIDE)       // 2304 floats

// ---------------------------------------------------------------------------
// CDNA5 async global->LDS (ASYNCcnt-tracked) with sync fallback
// builtin signature (from clang diagnostic): (global intN*, local intN*, Ii, Ii)
// ---------------------------------------------------------------------------
typedef __attribute__((address_space(1))) v4i gv4i;
typedef __attribute__((address_space(3))) v4i lv4i;
typedef __attribute__((address_space(1))) v2i gv2i;
typedef __attribute__((address_space(3))) v2i lv2i;

#if __has_builtin(__builtin_amdgcn_global_load_async_to_lds_b128) && \
    __has_builtin(__builtin_amdgcn_global_load_async_to_lds_b64)
#define HAVE_ASYNC_LDS 1
#else
#define HAVE_ASYNC_LDS 0
#endif

#if __has_builtin(__builtin_amdgcn_s_wait_asynccnt)
#define WAIT_ASYNC(n) __builtin_amdgcn_s_wait_asynccnt(n)
#else
#define WAIT_ASYNC(n) asm volatile("s_wait_asynccnt %0" ::"n"(n))
#endif

#if HAVE_ASYNC_LDS
__device__ __forceinline__ void async_b128(const float* g, float* l) {
  __builtin_amdgcn_global_load_async_to_lds_b128((gv4i*)g, (lv4i*)l, 0, 0);
}
__device__ __forceinline__ void async_b64(const float* g, float* l) {
  __builtin_amdgcn_global_load_async_to_lds_b64((gv2i*)g, (lv2i*)l, 0, 0);
}
#endif

// ---------------------------------------------------------------------------
// d[i] = D[i,i]
// ---------------------------------------------------------------------------
__global__ __launch_bounds__(256) void cotrec_diag(const float* __restrict__ D,
                                                   float* __restrict__ d) {
  int i = blockIdx.x * 256 + threadIdx.x;
  if (i < BDIM) d[i] = D[(size_t)i * (BDIM + 1)];
}

// ---------------------------------------------------------------------------
// Wt[l][k][n] = W[l][n][k]  (so the GEMM always consumes contiguous B rows)
// ---------------------------------------------------------------------------
__global__ __launch_bounds__(256) void cotrec_transpose_w(
    const float* __restrict__ W, float* __restrict__ Wt) {
  int i = blockIdx.x * 256 + threadIdx.x;
  if (i < LAYERS * EE * EE) {
    int l = i / (EE * EE);
    int rem = i - l * EE * EE;
    int k = rem / EE, n = rem - k * EE;
    Wt[(size_t)l * EE * EE + k * EE + n] = W[(size_t)l * EE * EE + n * EE + k];
  }
}

// ---------------------------------------------------------------------------
// X0[b,:] = (sum_l emb[session_item[b,l]-1]) / session_len[b]; idx==0 -> PAD
// one wave32 per session; lanes 0..27 each own one float4
// ---------------------------------------------------------------------------
__global__ __launch_bounds__(256) void cotrec_gather_pool(
    const float* __restrict__ emb, const int* __restrict__ items,
    const float* __restrict__ slen, float* __restrict__ X,
    float* __restrict__ acc) {
  const int wave = threadIdx.x >> 5;
  const int lane = threadIdx.x & 31;
  const int b = blockIdx.x * 8 + wave;
  const int* it = items + (size_t)b * MAXLEN;
  float4 s = make_float4(0.f, 0.f, 0.f, 0.f);
  for (int l = 0; l < MAXLEN; ++l) {
    int idx = it[l];  // wave-uniform -> scalarized
    if (idx > 0 && lane < 28) {
      const float4 v = *(const float4*)(emb + (size_t)(idx - 1) * EE + lane * 4);
      s.x += v.x; s.y += v.y; s.z += v.z; s.w += v.w;
    }
  }
  if (lane < 28) {
    const float inv = 1.0f / slen[b];
    float4 r = make_float4(s.x * inv, s.y * inv, s.z * inv, s.w * inv);
    *(float4*)(X + (size_t)b * EE + lane * 4) = r;
    *(float4*)(acc + (size_t)b * EE + lane * 4) = r;
  }
}

// ---------------------------------------------------------------------------
// C[4096,112] = rowScale ⊙ (A[4096,K] @ B[K,112])   fp32 WMMA 16x16x4
// block = 128 thr = 4 wave32: wave = (strip 0..1) x (nhalf 0..1)
//   strip -> 16-row band; nhalf -> N-tiles {0..3} or {4..7} (tile 7 = ghost)
// double-buffered LDS, CDNA5 async global->LDS, 8 async instr/wave/chunk
// ---------------------------------------------------------------------------
__device__ __forceinline__ void cotrec_issue_tiles(
    const float* __restrict__ A, int lda, const float* __restrict__ B, int kb,
    int row0, int t, float* AsBuf, float* BsBuf) {
#if HAVE_ASYNC_LDS
  // A tile: 32 rows x 16 k = 128 float4 -> exactly 1 b128 per thread
  {
    int r = t >> 2, kq = t & 3;
    async_b128(A + (size_t)(row0 + r) * lda + kb + kq * 4,
               AsBuf + r * ASTRIDE + kq * 4);
  }
  // B tile: 16 rows x 112 = 896 float2 -> exactly 7 b64 per thread
#pragma unroll
  for (int j = 0; j < 7; ++j) {
    int idx = t + j * 128;  // 0..895
    int kk = idx / 56, c2 = idx - kk * 56;
    async_b64(B + (size_t)(kb + kk) * EE + c2 * 2,
              BsBuf + kk * BSTRIDE + c2 * 2);
  }
#else
  for (int idx = t; idx < TMROWS * 16; idx += 128) {
    int r = idx >> 4, kk = idx & 15;
    AsBuf[r * ASTRIDE + kk] = A[(size_t)(row0 + r) * lda + kb + kk];
  }
  for (int idx = t; idx < 16 * EE; idx += 128) {
    int kk = idx / EE, col = idx - kk * EE;
    BsBuf[kk * BSTRIDE + col] = B[(size_t)(kb + kk) * EE + col];
  }
#endif
}

__global__ __launch_bounds__(128) void cotrec_gemm_wmma(
    const float* __restrict__ A, int lda, const float* __restrict__ B,
    const float* __restrict__ rowScale, float* __restrict__ C, int K) {
  __shared__ float As[2][A_ELEMS];
  __shared__ float Bs[2][B_ELEMS];
  const int t = threadIdx.x;
  const int wave = t >> 5;
  const int lane = t & 31;
  const int half = lane >> 4;   // K-pair selector (fp32 WMMA lane layout)
  const int l16 = lane & 15;    // M (A frag) / N (B frag) within a tile
  const int strip = wave >> 1;  // 0..1 -> 16-row band
  const int nhalf = wave & 1;   // 0..1 -> tiles {0..3} / {4..7}
  const int row0 = blockIdx.x * TMROWS;

  v8f acc[4] = {};

  const int nc = K >> 4;  // K is a multiple of 16 (112 or 4096)
  cotrec_issue_tiles(A, lda, B, 0, row0, t, As[0], Bs[0]);
  for (int c = 0; c < nc; ++c) {
    const int cur = c & 1;
    if (c + 1 < nc) {
      cotrec_issue_tiles(A, lda, B, (c + 1) << 4, row0, t, As[1 - cur],
                         Bs[1 - cur]);
#if HAVE_ASYNC_LDS
      WAIT_ASYNC(8);  // chunk c's 8 async ops done; chunk c+1 in flight
#endif
    } else {
#if HAVE_ASYNC_LDS
      WAIT_ASYNC(0);
#endif
    }
    __syncthreads();  // all waves' tile-c data visible in LDS

    const float* Asb = As[cur];
    const float* Bsb = Bs[cur];
#pragma unroll
    for (int ks = 0; ks < 4; ++ks) {
      const int k = ks * 4 + half * 2;
      v2f a;
      a.x = Asb[(strip * 16 + l16) * ASTRIDE + k];
      a.y = Asb[(strip * 16 + l16) * ASTRIDE + k + 1];
#pragma unroll
      for (int n = 0; n < 4; ++n) {  // tile 7 is a ghost (discarded)
        const int col = (nhalf * 4 + n) * 16 + l16;
        v2f bfr;
        bfr.x = Bsb[k * BSTRIDE + col];
        bfr.y = Bsb[(k + 1) * BSTRIDE + col];
        acc[n] = __builtin_amdgcn_wmma_f32_16x16x4_f32(
            false, a, false, bfr, (short)0, acc[n], false, false);
      }
    }
    __syncthreads();  // tile-c buffer free for chunk c+2's async writes
  }

  // epilogue: VGPR v holds row M = v + 8*half, col = l16 (16x16 f32 D layout)
#pragma unroll
  for (int n = 0; n < 4; ++n) {
    const int tile = nhalf * 4 + n;
    if (tile < 7) {
#pragma unroll
      for (int v = 0; v < 8; ++v) {
        const int row = row0 + strip * 16 + v + 8 * half;
        const float s = rowScale ? rowScale[row] : 1.0f;
        C[(size_t)row * EE + tile * 16 + l16] = acc[n][v] * s;
      }
    }
  }
}

// ---------------------------------------------------------------------------
// acc[b,:] += Y[b,:] / max(||Y[b,:]||, 1e-12)  — one wave32 per row
// ---------------------------------------------------------------------------
__global__ __launch_bounds__(256) void cotrec_norm_acc(
    const float* __restrict__ Y, float* __restrict__ acc) {
  const int wave = threadIdx.x >> 5;
  const int lane = threadIdx.x & 31;
  const int b = blockIdx.x * 8 + wave;
  float4 y = make_float4(0.f, 0.f, 0.f, 0.f);
  if (lane < 28) y = *(const float4*)(Y + (size_t)b * EE + lane * 4);
  float ss = y.x * y.x + y.y * y.y + y.z * y.z + y.w * y.w;
#pragma unroll
  for (int off = 16; off > 0; off >>= 1) ss += __shfl_xor(ss, off, 32);
  const float inv = 1.0f / fmaxf(sqrtf(ss), 1e-12f);
  if (lane < 28) {
    float4 a = *(float4*)(acc + (size_t)b * EE + lane * 4);
    a.x += y.x * inv; a.y += y.y * inv; a.z += y.z * inv; a.w += y.w * inv;
    *(float4*)(acc + (size_t)b * EE + lane * 4) = a;
  }
}

// ---------------------------------------------------------------------------
// out = acc / (LAYERS + 1)
// ---------------------------------------------------------------------------
__global__ __launch_bounds__(256) void cotrec_final(
    const float* __restrict__ acc, float* __restrict__ out, int n) {
  int i = blockIdx.x * 256 + threadIdx.x;
  if (i < n) out[i] = acc[i] * (1.0f / (LAYERS + 1));
}

// ---------------------------------------------------------------------------
extern "C" void kernel_launch(void* const* d_in, const int* in_sizes, int n_in,
                              void* d_out, int out_size, void* d_ws,
                              size_t ws_size, hipStream_t stream) {
  const float* emb   = (const float*)d_in[0];  // [N_ITEMS, 112]
  const float* Dm    = (const float*)d_in[1];  // [4096, 4096]
  const float* Am    = (const float*)d_in[2];  // [4096, 4096]
  const float* slen  = (const float*)d_in[3];  // [4096, 1]
  const float* Ws    = (const float*)d_in[4];  // [3, 112, 112]
  const int*   items = (const int*)d_in[5];    // [4096, 200]
  float* out = (float*)d_out;                  // [4096, 112]

  const size_t BE = (size_t)BDIM * EE;  // 458752
  float* ws   = (float*)d_ws;
  float* X    = ws;             // session_emb (ping)
  float* T    = X + BE;         // X @ W^T
  float* Y    = T + BE;         // d ⊙ (A @ T)  (pong)
  float* accb = Y + BE;         // running sum of terms
  float* dvec = accb + BE;      // diag(D)                [4096]
  float* Wt   = dvec + BDIM;    // transposed weights     [3,112,112]

  cotrec_diag<<<(BDIM + 255) / 256, 256, 0, stream>>>(Dm, dvec);
  cotrec_transpose_w<<<(LAYERS * EE * EE + 255) / 256, 256, 0, stream>>>(Ws,
                                                                         Wt);
  cotrec_gather_pool<<<BDIM / 8, 256, 0, stream>>>(emb, items, slen, X, accb);

  float* cur = X;
  float* nxt = Y;
  for (int l = 0; l < LAYERS; ++l) {
    // T = cur @ W_l^T            (M=4096, N=112, K=112; B = Wt_l is K-major)
    cotrec_gemm_wmma<<<BDIM / TMROWS, 128, 0, stream>>>(
        cur, EE, Wt + (size_t)l * EE * EE, nullptr, T, EE);
    // nxt = d ⊙ (A @ T)          (M=4096, N=112, K=4096)
    cotrec_gemm_wmma<<<BDIM / TMROWS, 128, 0, stream>>>(Am, BDIM, T, dvec, nxt,
                                                        BDIM);
    // accb += normalize(nxt); next layer consumes UN-normalized nxt
    cotrec_norm_acc<<<BDIM / 8, 256, 0, stream>>>(nxt, accb);
    float* tmp = cur; cur = nxt; nxt = tmp;
  }

  cotrec_final<<<((int)BE + 255) / 256, 256, 0, stream>>>(accb, out, (int)BE);
}